// Fast2commMultiHead_55130200211607
// MI455X (gfx1250) — compile-verified
//
#include <hip/hip_runtime.h>
#include <hip/hip_bf16.h>
#include <math.h>

typedef __attribute__((ext_vector_type(2))) float v2f;
typedef __attribute__((ext_vector_type(8))) float v8f;

namespace {
constexpr int Bc = 2, Lc = 5, Hc = 96, Wc = 352, Cc = 64;
constexpr int HWc = Hc * Wc;        // 33792
constexpr int BLc = Bc * Lc;        // 10
constexpr int NBOX = 8;
constexpr int XT = Wc / 16;         // 22 x-tiles of 16 pixels
}

// ---------- helpers ----------
__device__ __forceinline__ unsigned f2u_ord(float f) {
    unsigned u = __float_as_uint(f);
    return u ^ ((u >> 31) ? 0xFFFFFFFFu : 0x80000000u);  // order-preserving map
}

__device__ __forceinline__ float conv_tap(const float* __restrict__ src, int y, int xcol, int k) {
    int kk = k < 25 ? k : 0;
    int ky = kk / 5 - 2;
    int kx = kk % 5 - 2;
    int yy = y + ky, xx = xcol + kx;
    bool inb = (k < 25) && (yy >= 0) && (yy < Hc) && (xx >= 0) && (xx < Wc);
    int yc = min(max(yy, 0), Hc - 1);
    int xc = min(max(xx, 0), Wc - 1);
    // unconditional clamped load + arithmetic predication: EXEC stays all-1s for WMMA
    return src[yc * Wc + xc] * (inb ? 1.0f : 0.0f);
}

__device__ __forceinline__ void softmax5(const float d[5], const float m[5], float w[5]) {
    float s[5], mx = -1e30f;
#pragma unroll
    for (int i = 0; i < 5; ++i) { s[i] = m[i] * d[i] * 0.125f; mx = fmaxf(mx, s[i]); }
    float sum = 0.f;
#pragma unroll
    for (int i = 0; i < 5; ++i) { s[i] = __expf(s[i] - mx); sum += s[i]; }
    float inv = 1.0f / sum;
#pragma unroll
    for (int i = 0; i < 5; ++i) w[i] = s[i] * inv * m[i];  // fold mask into combine weight
}

// ---------- 1) conf = sigmoid(max over anchors); also zero gt counter ----------
__global__ void k_conf(const float* __restrict__ cm, float* __restrict__ conf,
                       unsigned* __restrict__ gt_cnt) {
    int i = blockIdx.x * blockDim.x + threadIdx.x;
    if (i == 0) *gt_cnt = 0u;
    if (i >= BLc * HWc) return;
    int g = i / HWc, p = i % HWc;                       // g = b*L + l
    float v0 = cm[(size_t)(g * 2 + 0) * HWc + p];
    float v1 = cm[(size_t)(g * 2 + 1) * HWc + p];
    float v = fmaxf(v0, v1);                            // sigmoid monotone: max then sigmoid
    conf[i] = 1.0f / (1.0f + __expf(-v));
}

// ---------- 2) 5x5 Gaussian conv via V_WMMA_F32_16X16X4_F32 (im2col, fp32 exact) ----------
__global__ void k_gauss_wmma(const float* __restrict__ conf, const float* __restrict__ gk,
                             float* __restrict__ smooth) {
    int wid = blockIdx.x;                 // one 32-lane wave per block
    int xt  = wid % XT;
    int y   = (wid / XT) % Hc;
    int img = wid / (XT * Hc);
    int lane = threadIdx.x;
    int half = lane >> 4;                 // K split across lane halves (A 16x4 layout)
    int lr   = lane & 15;                 // M (pixel within tile)
    int x0   = xt * 16;
    const float* src = conf + (size_t)img * HWc;

    v8f acc = {};
#pragma unroll
    for (int k0 = 0; k0 < 28; k0 += 4) {  // 25 taps padded to 28 = 7 K-steps of 4
        int ka0 = k0 + half * 2;
        int ka1 = ka0 + 1;
        v2f a, b;
        a.x = conv_tap(src, y, x0 + lr, ka0);
        a.y = conv_tap(src, y, x0 + lr, ka1);
        int kb0 = min(ka0, 24), kb1 = min(ka1, 24);
        b.x = gk[kb0] * ((ka0 < 25) ? 1.0f : 0.0f);     // kernel tap broadcast over N
        b.y = gk[kb1] * ((ka1 < 25) ? 1.0f : 0.0f);
        acc = __builtin_amdgcn_wmma_f32_16x16x4_f32(false, a, false, b, (short)0, acc,
                                                    false, false);
    }
    // D[m][n] = smooth(pixel m) for every n. Lane 0 holds M=0..7, lane 16 holds M=8..15.
    if (lr == 0) {
        float* dst = smooth + (size_t)img * HWc + (size_t)y * Wc + x0 + half * 8;
#pragma unroll
        for (int r = 0; r < 8; ++r) dst[r] = acc[r];
    }
}

// ---------- 3) exact top-K threshold per (b,l): 4-pass MSB radix select ----------
__global__ void k_radix_select(const float* __restrict__ smooth, const int* __restrict__ Kp,
                               unsigned* __restrict__ thr_u, unsigned* __restrict__ thr_r,
                               unsigned* __restrict__ eq_cnt) {
    __shared__ unsigned hist[256];
    __shared__ unsigned s_prefix, s_remaining;
    int g = blockIdx.x;                                   // 0..BL-1
    const float* data = smooth + (size_t)g * HWc;
    int tid = threadIdx.x;                                // 256 threads
    if (tid == 0) { s_prefix = 0u; s_remaining = (unsigned)(*Kp); eq_cnt[g] = 0u; }
    __syncthreads();
    for (int pass = 0; pass < 4; ++pass) {
        int shift = 24 - 8 * pass;
        unsigned prefmask = (pass == 0) ? 0u : (0xFFFFFFFFu << (shift + 8));
        hist[tid] = 0u;
        __syncthreads();
        unsigned prefix = s_prefix;
        for (int i = tid; i < HWc; i += 256) {
            unsigned u = f2u_ord(data[i]);
            if ((u & prefmask) == (prefix & prefmask))
                atomicAdd(&hist[(u >> shift) & 255u], 1u);
        }
        __syncthreads();
        if (tid == 0) {
            unsigned rem = s_remaining, cum = 0u;
            int bin = 0;
            for (int t = 255; t >= 0; --t) {
                if (cum + hist[t] >= rem) { bin = t; break; }
                cum += hist[t];
            }
            s_remaining = rem - cum;
            s_prefix = prefix | ((unsigned)bin << shift);
        }
        __syncthreads();
    }
    if (tid == 0) { thr_u[g] = s_prefix; thr_r[g] = s_remaining; }
}

// ---------- 4) dense top-K mask (exactly K ones per (b,l)) ----------
__global__ void k_build_mask(const float* __restrict__ smooth, const unsigned* __restrict__ thr_u,
                             const unsigned* __restrict__ thr_r, unsigned* __restrict__ eq_cnt,
                             float* __restrict__ mask_conf) {
    int i = blockIdx.x * blockDim.x + threadIdx.x;
    if (i >= BLc * HWc) return;
    int g = i / HWc;
    unsigned u = f2u_ord(smooth[i]);
    unsigned t = thr_u[g];
    float m = 0.0f;
    if (u > t) m = 1.0f;
    else if (u == t) {
        unsigned slot = atomicAdd(&eq_cnt[g], 1u);
        if (slot < thr_r[g]) m = 1.0f;                    // admit exactly the needed tie count
    }
    mask_conf[i] = m;
}

// ---------- 5) GT-box union mask + global count ----------
__global__ void k_gt_mask(const int* __restrict__ gt_boxes, float* __restrict__ mask_gt,
                          unsigned* __restrict__ gt_cnt) {
    __shared__ unsigned cnt;
    if (threadIdx.x == 0) cnt = 0u;
    __syncthreads();
    int i = blockIdx.x * blockDim.x + threadIdx.x;        // over B*HW
    if (i < Bc * HWc) {
        int b = i / HWc, p = i % HWc;
        int y = p / Wc, x = p % Wc;
        float m = 0.0f;
        for (int j = 0; j < NBOX; ++j) {
            const int* bx = gt_boxes + (b * NBOX + j) * 4;
            int x1 = min(max(bx[0], 0), Wc), y1 = min(max(bx[1], 0), Hc);
            int x2 = min(max(bx[2], 0), Wc), y2 = min(max(bx[3], 0), Hc);
            if (y >= y1 && y < y2 && x >= x1 && x < x2) { m = 1.0f; break; }
        }
        mask_gt[i] = m;
        if (m > 0.0f) atomicAdd(&cnt, 1u);
    }
    __syncthreads();
    if (threadIdx.x == 0 && cnt) atomicAdd(gt_cnt, cnt);
}

// ---------- 6) rate = K/HW + gt_count/(B*HW) (mask_conf has exactly K ones per (b,l)) ----------
__global__ void k_rate(const int* __restrict__ Kp, const unsigned* __restrict__ gt_cnt,
                       float* __restrict__ out_rate) {
    *out_rate = (float)(*Kp) / (float)HWc + (float)(*gt_cnt) / (float)(Bc * HWc);
}

// ---------- 7) fused dual attention (conf + gt share the raw agent-0 dots) ----------
__global__ void k_fuse(const float* __restrict__ x, const float* __restrict__ mask_conf,
                       const float* __restrict__ mask_gt, float* __restrict__ out) {
    int t = blockIdx.x * blockDim.x + threadIdx.x;        // one thread per (b, pixel)
    if (t >= Bc * HWc) return;
    int b = t / HWc, p = t % HWc;
    const float* xb = x + (size_t)b * Lc * Cc * HWc + p;

    // pass 1: raw dots d[m] = <q0, qm> over C (coalesced: lanes span consecutive p)
    float d0 = 0, d1 = 0, d2 = 0, d3 = 0, d4 = 0;
    for (int c = 0; c < Cc; ++c) {
        float q0 = xb[(size_t)(0 * Cc + c) * HWc];
        float q1 = xb[(size_t)(1 * Cc + c) * HWc];
        float q2 = xb[(size_t)(2 * Cc + c) * HWc];
        float q3 = xb[(size_t)(3 * Cc + c) * HWc];
        float q4 = xb[(size_t)(4 * Cc + c) * HWc];
        d0 = fmaf(q0, q0, d0); d1 = fmaf(q0, q1, d1); d2 = fmaf(q0, q2, d2);
        d3 = fmaf(q0, q3, d3); d4 = fmaf(q0, q4, d4);
    }
    float d[5] = {d0, d1, d2, d3, d4};
    float mc[5], mg[5];
    mc[0] = 1.0f; mg[0] = 1.0f;                           // ego forced to 1 (after rate)
    float gtv = mask_gt[t];
#pragma unroll
    for (int m = 1; m < 5; ++m) {
        mc[m] = mask_conf[(size_t)(b * Lc + m) * HWc + p];
        mg[m] = gtv;
    }
    float wc[5], wg[5];
    softmax5(d, mc, wc);
    softmax5(d, mg, wg);

    // pass 2: out[c] = sum_m w[m]*mask[m]*q[m][c]  (x re-read hits L2: 86.5MB < 192MB)
    float* oc = out + (size_t)b * Cc * HWc + p;
    float* og = oc + (size_t)Bc * Cc * HWc;
    for (int c = 0; c < Cc; ++c) {
        float q0 = xb[(size_t)(0 * Cc + c) * HWc];
        float q1 = xb[(size_t)(1 * Cc + c) * HWc];
        float q2 = xb[(size_t)(2 * Cc + c) * HWc];
        float q3 = xb[(size_t)(3 * Cc + c) * HWc];
        float q4 = xb[(size_t)(4 * Cc + c) * HWc];
        float vc = wc[0]*q0 + wc[1]*q1 + wc[2]*q2 + wc[3]*q3 + wc[4]*q4;
        float vg = wg[0]*q0 + wg[1]*q1 + wg[2]*q2 + wg[3]*q3 + wg[4]*q4;
        oc[(size_t)c * HWc] = vc;
        og[(size_t)c * HWc] = vg;
    }
}

extern "C" void kernel_launch(void* const* d_in, const int* in_sizes, int n_in,
                              void* d_out, int out_size, void* d_ws, size_t ws_size,
                              hipStream_t stream) {
    const float* x   = (const float*)d_in[0];             // (B*L, C, H, W)
    const float* cm  = (const float*)d_in[1];             // (B, L, A, H, W)
    const float* gk  = (const float*)d_in[2];             // (1,1,5,5) -> 25 floats
    const int*   gtb = (const int*)d_in[3];               // (B, 8, 4)
    const int*   Kp  = (const int*)d_in[4];               // scalar K
    float* out = (float*)d_out;                           // [fused_conf | fused_gt | rate]

    // workspace layout (~4.3 MB)
    float* conf      = (float*)d_ws;
    float* smooth    = conf + (size_t)BLc * HWc;
    float* mask_conf = smooth + (size_t)BLc * HWc;
    float* mask_gt   = mask_conf + (size_t)BLc * HWc;
    unsigned* thr_u  = (unsigned*)(mask_gt + (size_t)Bc * HWc);
    unsigned* thr_r  = thr_u + BLc;
    unsigned* eq_cnt = thr_r + BLc;
    unsigned* gt_cnt = eq_cnt + BLc;

    const int N_blhw = BLc * HWc;                         // 337920
    const int N_bhw  = Bc * HWc;                          // 67584

    k_conf<<<(N_blhw + 255) / 256, 256, 0, stream>>>(cm, conf, gt_cnt);
    k_gauss_wmma<<<BLc * Hc * XT, 32, 0, stream>>>(conf, gk, smooth);
    k_radix_select<<<BLc, 256, 0, stream>>>(smooth, Kp, thr_u, thr_r, eq_cnt);
    k_build_mask<<<(N_blhw + 255) / 256, 256, 0, stream>>>(smooth, thr_u, thr_r, eq_cnt, mask_conf);
    k_gt_mask<<<(N_bhw + 255) / 256, 256, 0, stream>>>(gtb, mask_gt, gt_cnt);
    k_rate<<<1, 1, 0, stream>>>(Kp, gt_cnt, out + (size_t)2 * Bc * Cc * HWc);
    k_fuse<<<(N_bhw + 255) / 256, 256, 0, stream>>>(x, mask_conf, mask_gt, out);
}